// CenterLoss_77378130805013
// MI455X (gfx1250) — compile-verified
//
#include <hip/hip_runtime.h>

#define NUM_CLASSES 8192
#define FEAT_DIM    256
#define BATCH       4096

typedef __attribute__((ext_vector_type(2))) float v2f;
typedef __attribute__((ext_vector_type(8))) float v8f;

// ---------------------------------------------------------------------------
// Stage 1: per-sample squared distance to its own class center.
// One wave (32 lanes) per sample; lanes stride float4 -> fully coalesced
// 512B/instruction global_load_b128 for both x row and gathered center row.
// 8 MB total traffic -> ~0.35 us at 23.3 TB/s; VALU work is negligible.
// ---------------------------------------------------------------------------
__global__ void center_dist_stage1(const float* __restrict__ x,
                                   const int*   __restrict__ labels,
                                   const float* __restrict__ centers,
                                   float*       __restrict__ dists) {
    const int wave = (int)((blockIdx.x * blockDim.x + threadIdx.x) >> 5);
    const int lane = (int)(threadIdx.x & 31);
    if (wave >= BATCH) return;

    const int b   = wave;
    const int lbl = labels[b];

    const float4* __restrict__ xr = (const float4*)(x       + (size_t)b   * FEAT_DIM);
    const float4* __restrict__ cr = (const float4*)(centers + (size_t)lbl * FEAT_DIM);

    float acc = 0.0f;
    // 256 floats = 64 float4 per row; 32 lanes -> 2 iterations.
#pragma unroll
    for (int i = 0; i < 2; ++i) {
        const int idx = lane + i * 32;
        const float4 xv = xr[idx];
        const float4 cv = cr[idx];
        const float d0 = xv.x - cv.x;
        const float d1 = xv.y - cv.y;
        const float d2 = xv.z - cv.z;
        const float d3 = xv.w - cv.w;
        acc = fmaf(d0, d0, acc);
        acc = fmaf(d1, d1, acc);
        acc = fmaf(d2, d2, acc);
        acc = fmaf(d3, d3, acc);
    }

    // wave32 butterfly reduction (fixed order -> deterministic)
#pragma unroll
    for (int off = 16; off > 0; off >>= 1)
        acc += __shfl_xor(acc, off, 32);

    if (lane == 0) dists[b] = acc;
}

// ---------------------------------------------------------------------------
// Stage 2: reduce the 4096 distances with V_WMMA_F32_16X16X4_F32.
// Single wave, EXEC all-1s (WMMA requirement). Each iteration loads 64
// contiguous floats as a 16x4 A tile (lane l holds A[l%16][2*(l/16)+{0,1}]
// -- a bijection, so the sum is layout-independent), clamps them per the
// reference, and accumulates D[m][n] += sum_k A[m][k] via B = ones(4x16).
// Grand total = sum_m D[m][0] = (lane0: sum of 8 D regs, rows 0..7) +
// (lane16: sum of 8 D regs, rows 8..15). Fixed order -> deterministic.
// ---------------------------------------------------------------------------
__global__ void center_reduce_stage2(const float* __restrict__ dists,
                                     float*       __restrict__ out) {
    const int lane = (int)threadIdx.x; // 0..31, exactly one wave

    v2f ones;
    ones.x = 1.0f;
    ones.y = 1.0f;

    v8f c = {};

    const float2* __restrict__ p2 = (const float2*)dists;

    for (int t = 0; t < BATCH / 64; ++t) {
        const float2 p = p2[t * 32 + lane];
        v2f a;
        a.x = fminf(fmaxf(p.x, 1e-12f), 1e12f);
        a.y = fminf(fmaxf(p.y, 1e-12f), 1e12f);
        // D = A x Ones + C : folds 64 clamped values into the accumulator.
        c = __builtin_amdgcn_wmma_f32_16x16x4_f32(
                /*neg_a=*/false, a,
                /*neg_b=*/false, ones,
                /*c_mod=*/(short)0, c,
                /*reuse_a=*/false, /*reuse_b=*/false);
    }

    float s = c[0] + c[1] + c[2] + c[3] + c[4] + c[5] + c[6] + c[7];

    // total = rows 0..7 (held in lane 0) + rows 8..15 (held in lane 16)
    const float tot = __shfl(s, 0, 32) + __shfl(s, 16, 32);

    if (lane == 0) {
        // + B*(C-1)*1e-12 from the clamped zero entries, then / B.
        out[0] = tot / (float)BATCH + (float)(NUM_CLASSES - 1) * 1e-12f;
    }
}

extern "C" void kernel_launch(void* const* d_in, const int* in_sizes, int n_in,
                              void* d_out, int out_size, void* d_ws, size_t ws_size,
                              hipStream_t stream) {
    (void)in_sizes; (void)n_in; (void)out_size; (void)ws_size;

    const float* x       = (const float*)d_in[0];
    const int*   labels  = (const int*)d_in[1];   // JAX default x64-off -> int32
    const float* centers = (const float*)d_in[2];
    float*       out     = (float*)d_out;
    float*       ws      = (float*)d_ws;          // 4096 floats = 16 KB

    // 8 waves per block (256 threads) -> 8 samples per block -> 512 blocks.
    center_dist_stage1<<<BATCH / 8, 256, 0, stream>>>(x, labels, centers, ws);
    center_reduce_stage2<<<1, 32, 0, stream>>>(ws, out);
}